// MDTA_24120536335154
// MI455X (gfx1250) — compile-verified
//
#include <hip/hip_runtime.h>

typedef float v2f __attribute__((ext_vector_type(2)));
typedef float v4f __attribute__((ext_vector_type(4)));
typedef float v8f __attribute__((ext_vector_type(8)));
typedef unsigned int v4u __attribute__((ext_vector_type(4)));
typedef int v4i __attribute__((ext_vector_type(4)));
typedef int v8i __attribute__((ext_vector_type(8)));

#define B_   4
#define C_   192
#define NH_  4
#define D_   48
#define HW_  256
#define NSP  65536L

#if __has_builtin(__builtin_amdgcn_tensor_load_to_lds)
#define USE_TDM 1
#else
#define USE_TDM 0
#endif

__device__ __forceinline__ float gelu_exact(float x) {
    return 0.5f * x * (1.0f + erff(x * 0.70710678118654752f));
}

#if USE_TDM
// ---------------------------------------------------------------------------
// TDM: DMA one 16(rows) x 128(cols) fp32 tile, row stride 65536 elements,
// into LDS with 8-DWORD padding after every 128 DWORDs (-> [16][136] layout).
// D# built per CDNA5 ISA 8.3/8.4: group0 = {count|flags, lds_addr, gaddr_lo,
// gaddr_hi|type=2}; group1 = {data_size=4B|pad cfg, tensor dims (huge),
// tile_dim0=128, tile_dim1=16, dim0_stride=65536}. Groups 2/3 zero (2-D
// tensor); trailing int32x8 of the 6-arg builtin zero-filled.
// ---------------------------------------------------------------------------
__device__ __forceinline__ void tdm_load_tile_16x128(unsigned lds_byte,
                                                     const void* gaddr)
{
    unsigned long long ga = (unsigned long long)gaddr;
    v4u g0;
    g0[0] = 1u;                                    // count=1, user mode
    g0[1] = lds_byte;                              // lds_addr (bytes)
    g0[2] = (unsigned)ga;                          // global_addr[31:0]
    g0[3] = ((unsigned)(ga >> 32) & 0x01FFFFFFu)   // global_addr[56:32]
            | (2u << 30);                          // type = 2 ("image")
    v8i g1;
    g1[0] = (int)((2u << 16)      // data_size = 4 bytes
                | (1u << 20)      // pad_enable
                | (6u << 22)      // pad_interval: 128 DWORDs
                | (7u << 25));    // pad_amount: 8 DWORDs
    g1[1] = (int)0xFFFF0000u;                     // tensor_dim0 lo16 (huge)
    g1[2] = (int)(0x3FFFu | 0xFFFF0000u);         // tensor_dim0 hi / dim1 lo
    g1[3] = (int)(0x3FFFu | (128u << 16));        // dim1 hi / tile_dim0=128
    g1[4] = 16;                                   // tile_dim1=16, tile_dim2=0
    g1[5] = 65536;                                // tensor_dim0_stride (elems)
    g1[6] = 0;
    g1[7] = 0;
    v4i gz4 = {0, 0, 0, 0};
    v8i gz8 = {0, 0, 0, 0, 0, 0, 0, 0};
    __builtin_amdgcn_tensor_load_to_lds(g0, g1, gz4, gz4, gz8, 0);
}
#endif

// ---------------------------------------------------------------------------
// Generic WMMA GEMM:  Y[g][m][n] = sum_k A[g][m][k] * X[g][k][n]   (+ GELU)
// group g -> (b = g/GH, h = g%GH) with independent strides for A, X, Y.
// Block: 128 threads (4 waves). Each wave computes a 16(M) x 128(N) strip;
// block tile 64 x 128. X tiles staged to LDS by the Tensor Data Mover
// (double buffered, TENSORcnt-pipelined); fallback: b128 loads + ds stores.
// Matrix math: V_WMMA_F32_16X16X4_F32 (full fp32, fp32 accumulate).
// ---------------------------------------------------------------------------
__global__ __launch_bounds__(128) void gemm_wmma(
    float* __restrict__ Y, const float* __restrict__ A, const float* __restrict__ X,
    int M, int K,
    long aBS, long aHS, long xBS, long xHS, long yBS, long yHS,
    int GH, int transA, int act)
{
    __shared__ float sX[2][16][136];
    const int tid  = threadIdx.x;
    const int wave = tid >> 5;
    const int lane = tid & 31;
    const int g = blockIdx.z;
    const int b = g / GH, h = g % GH;
    const float* Ag = A + (long)b * aBS + (long)h * aHS;
    const float* Xg = X + (long)b * xBS + (long)h * xHS;
    float*       Yg = Y + (long)b * yBS + (long)h * yHS;
    const int  m0 = blockIdx.y * 64 + wave * 16;
    const long n0 = (long)blockIdx.x * 128;
    const bool active = (m0 < M);          // wave-uniform: EXEC stays full
    const int row   = lane & 15;           // M-row (A) / N-col (B)
    const int khalf = (lane >> 4) << 1;    // 0 (lanes 0-15) or 2 (lanes 16-31)

    v8f acc[8] = {};

#if USE_TDM
    const unsigned lds0 = (unsigned)(unsigned long long)&sX[0][0][0];
    const unsigned lds1 = (unsigned)(unsigned long long)&sX[1][0][0];
    const bool issuer = (wave == 0);       // TDM: one DMA per block per chunk
    const int nch = K >> 4;
    if (issuer)
        tdm_load_tile_16x128(lds0, Xg + n0);
    for (int ci = 0; ci < nch; ci++) {
        __syncthreads();                   // prior reads of the victim buffer done
        if (issuer) {
            if (ci + 1 < nch) {
                tdm_load_tile_16x128((ci & 1) ? lds0 : lds1,
                                     Xg + (long)((ci + 1) << 4) * NSP + n0);
                __builtin_amdgcn_s_wait_tensorcnt((short)1);  // chunk ci landed
            } else {
                __builtin_amdgcn_s_wait_tensorcnt((short)0);
            }
        }
        __syncthreads();                   // tile visible to all waves
        if (active) {
            const int k0 = ci << 4;
            const float (*sxb)[136] = sX[ci & 1];
            #pragma unroll
            for (int kk = 0; kk < 16; kk += 4) {
                const int ka = k0 + kk + khalf;
                v2f a;
                if (!transA) {
                    a.x = Ag[(long)(m0 + row) * K + ka];
                    a.y = Ag[(long)(m0 + row) * K + ka + 1];
                } else {      // A[m,k] stored as A[k*M + m]  (context^T case)
                    a.x = Ag[(long)ka * M + (m0 + row)];
                    a.y = Ag[(long)(ka + 1) * M + (m0 + row)];
                }
                #pragma unroll
                for (int j = 0; j < 8; j++) {
                    v2f bb;
                    bb.x = sxb[kk + khalf][j * 16 + row];
                    bb.y = sxb[kk + khalf + 1][j * 16 + row];
                    acc[j] = __builtin_amdgcn_wmma_f32_16x16x4_f32(
                        false, a, false, bb, (short)0, acc[j], false, false);
                }
            }
        }
    }
#else
    const int lr = tid >> 5;               // 0..3
    const int lc = (tid & 31) << 2;        // 0..124 step 4
    for (int k0 = 0; k0 < K; k0 += 16) {
        __syncthreads();
        #pragma unroll
        for (int q = 0; q < 4; q++) {
            int kr = lr + q * 4;
            v4f v = *(const v4f*)(Xg + (long)(k0 + kr) * NSP + n0 + lc);
            *(v4f*)(&sX[0][kr][lc]) = v;
        }
        if (k0 + 16 < K)
            __builtin_prefetch(Xg + (long)(k0 + 16 + lr) * NSP + n0 + lc, 0, 3);
        __syncthreads();
        if (active) {
            #pragma unroll
            for (int kk = 0; kk < 16; kk += 4) {
                const int ka = k0 + kk + khalf;
                v2f a;
                if (!transA) {
                    a.x = Ag[(long)(m0 + row) * K + ka];
                    a.y = Ag[(long)(m0 + row) * K + ka + 1];
                } else {
                    a.x = Ag[(long)ka * M + (m0 + row)];
                    a.y = Ag[(long)(ka + 1) * M + (m0 + row)];
                }
                #pragma unroll
                for (int j = 0; j < 8; j++) {
                    v2f bb;
                    bb.x = sX[0][kk + khalf][j * 16 + row];
                    bb.y = sX[0][kk + khalf + 1][j * 16 + row];
                    acc[j] = __builtin_amdgcn_wmma_f32_16x16x4_f32(
                        false, a, false, bb, (short)0, acc[j], false, false);
                }
            }
        }
    }
#endif

    if (active) {
        const int moff = (lane >= 16) ? 8 : 0;
        #pragma unroll
        for (int j = 0; j < 8; j++) {
            long nb = n0 + j * 16 + row;
            #pragma unroll
            for (int r = 0; r < 8; r++) {
                float v = acc[j][r];
                if (act == 1) v = gelu_exact(v);
                Yg[(long)(m0 + r + moff) * NSP + nb] = v;
            }
        }
    }
}

// ---------------------------------------------------------------------------
// Split-K WMMA reduction: Y[g][i][j] += sum_n Xa[g][i][n] * Xb[g][j][n]
// (the bhdn,bhen->bhde einsums, K = 65536).  One wave per block; fp32 WMMA
// over K chunks, partials combined with global_atomic_add_f32.
// ---------------------------------------------------------------------------
__global__ __launch_bounds__(32) void dotk_wmma(
    float* __restrict__ Yo, const float* __restrict__ Xa, const float* __restrict__ Xb,
    long aBS, long aHS, long bBS, long bHS, int GH, int nsplit)
{
    const int lane = threadIdx.x;
    const int gz = blockIdx.z;
    const int g = gz / nsplit, sp = gz % nsplit;
    const int b = g / GH, h = g % GH;
    const float* Am = Xa + (long)b * aBS + (long)h * aHS;
    const float* Bm = Xb + (long)b * bBS + (long)h * bHS;
    const long chunk = NSP / nsplit;
    const long nbeg  = (long)sp * chunk;
    const int i0 = blockIdx.x * 16, j0 = blockIdx.y * 16;
    const int row   = lane & 15;
    const int khalf = (lane >> 4) << 1;
    const float* pa = Am + (long)(i0 + row) * NSP + khalf;
    const float* pb = Bm + (long)(j0 + row) * NSP + khalf;
    v8f acc = {};
    #pragma unroll 4
    for (long n = nbeg; n < nbeg + chunk; n += 4) {
        v2f a  = *(const v2f*)(pa + n);
        v2f bb = *(const v2f*)(pb + n);
        acc = __builtin_amdgcn_wmma_f32_16x16x4_f32(
            false, a, false, bb, (short)0, acc, false, false);
    }
    float* Yg = Yo + (long)g * (D_ * D_);
    const int moff = (lane >= 16) ? 8 : 0;
    #pragma unroll
    for (int r = 0; r < 8; r++)
        atomicAdd(&Yg[(i0 + r + moff) * D_ + j0 + row], acc[r]);
}

// ---------------------------------------------------------------------------
// Online softmax over N=65536 for the q,k rows of qkv (in place).
// ---------------------------------------------------------------------------
__global__ __launch_bounds__(256) void softmax_rows_qk(float* __restrict__ qkv)
{
    __shared__ float sm[256], ss[256];
    const int r = blockIdx.x;
    const int b = r / (2 * C_), ch = r % (2 * C_);
    float* rowp = qkv + ((long)b * (3 * C_) + ch) * NSP;
    const int tid = threadIdx.x;
    float m = -3.402823466e38f, s = 0.f;
    for (long i = tid; i < NSP; i += 256) {
        float v = rowp[i];
        if (v > m) { s = s * __expf(m - v) + 1.f; m = v; }
        else       { s += __expf(v - m); }
    }
    sm[tid] = m; ss[tid] = s;
    __syncthreads();
    for (int st = 128; st > 0; st >>= 1) {
        if (tid < st) {
            float m2 = sm[tid + st], s2 = ss[tid + st];
            float M = fmaxf(sm[tid], m2);
            ss[tid] = ss[tid] * __expf(sm[tid] - M) + s2 * __expf(m2 - M);
            sm[tid] = M;
        }
        __syncthreads();
    }
    const float M = sm[0];
    const float inv = 1.f / ss[0];
    for (long i = tid; i < NSP; i += 256)
        rowp[i] = __expf(rowp[i] - M) * inv;
}

// Per-row L2 norm over N (clamped like F.normalize).
__global__ __launch_bounds__(256) void rownorm(
    const float* __restrict__ qf, float* __restrict__ norms)
{
    __shared__ float ss[256];
    const int r = blockIdx.x;
    const float* rowp = qf + (long)r * NSP;
    const int tid = threadIdx.x;
    float s = 0.f;
    for (long i = tid; i < NSP; i += 256) { float v = rowp[i]; s += v * v; }
    ss[tid] = s; __syncthreads();
    for (int st = 128; st > 0; st >>= 1) {
        if (tid < st) ss[tid] += ss[tid + st];
        __syncthreads();
    }
    if (tid == 0) norms[r] = fmaxf(sqrtf(ss[0]), 1e-12f);
}

// Tiny softmax over e=48: logits = S[d,e] * T[h] / ||qf_d||  (norm folded in).
__global__ __launch_bounds__(64) void attn_softmax(
    const float* __restrict__ S, const float* __restrict__ norms,
    const float* __restrict__ temp, float* __restrict__ attnf)
{
    __shared__ float sv[64];
    const int r = blockIdx.x;              // (b*NH+h)*48 + d
    const int h = (r / D_) % NH_;
    const int tid = threadIdx.x;
    const float scale = temp[h] / norms[r];
    float v = (tid < D_) ? S[(long)r * D_ + tid] * scale : -3.402823466e38f;
    sv[tid] = v; __syncthreads();
    for (int st = 32; st > 0; st >>= 1) {
        if (tid < st) sv[tid] = fmaxf(sv[tid], sv[tid + st]);
        __syncthreads();
    }
    const float M = sv[0]; __syncthreads();
    float e = (tid < D_) ? __expf(v - M) : 0.f;
    sv[tid] = e; __syncthreads();
    for (int st = 32; st > 0; st >>= 1) {
        if (tid < st) sv[tid] += sv[tid + st];
        __syncthreads();
    }
    if (tid < D_) attnf[(long)r * D_ + tid] = e / sv[0];
}

// ---------------------------------------------------------------------------
// 256-pt radix-2 DIT FFT, row pass: real input -> complex output.
// Block = 256 threads = 2 rows x 128 threads.
// ---------------------------------------------------------------------------
__global__ __launch_bounds__(256) void fft_rows(
    const float* __restrict__ in, float2* __restrict__ out)
{
    __shared__ float re[2][256], im[2][256];
    const int sub = threadIdx.x >> 7;
    const int t   = threadIdx.x & 127;
    const long rowid = (long)blockIdx.x * 2 + sub;
    const float* rp = in + rowid * 256;
    const int i0 = t, i1 = t + 128;
    const int r0 = __brev(i0) >> 24, r1 = __brev(i1) >> 24;
    re[sub][r0] = rp[i0]; im[sub][r0] = 0.f;
    re[sub][r1] = rp[i1]; im[sub][r1] = 0.f;
    __syncthreads();
    #pragma unroll
    for (int s = 1; s <= 8; s++) {
        const int half = 1 << (s - 1);
        const int blk = t >> (s - 1);
        const int pos = t & (half - 1);
        const int i = (blk << s) + pos;
        const int j = i + half;
        const float ang = -6.28318530717958647f * (float)pos / (float)(1 << s);
        const float wc = cosf(ang), wsn = sinf(ang);
        float xr = re[sub][j], xi = im[sub][j];
        float tr = wc * xr - wsn * xi;
        float ti = wc * xi + wsn * xr;
        float ur = re[sub][i], ui = im[sub][i];
        re[sub][i] = ur + tr; im[sub][i] = ui + ti;
        re[sub][j] = ur - tr; im[sub][j] = ui - ti;
        __syncthreads();
    }
    out[rowid * 256 + i0] = make_float2(re[sub][i0], im[sub][i0]);
    out[rowid * 256 + i1] = make_float2(re[sub][i1], im[sub][i1]);
}

// Column pass: complex input -> real part (x scale). 16 columns per block.
__global__ __launch_bounds__(256) void fft_cols(
    const float2* __restrict__ in, float* __restrict__ outr, float scale)
{
    __shared__ float re[256][16], im[256][16];
    const int img = blockIdx.x;
    const int c0  = blockIdx.y * 16;
    const int tid = threadIdx.x;
    const int cc  = tid & 15;
    const int trr = tid >> 4;             // 0..15
    const float2* base = in + (long)img * 65536 + c0;
    for (int r0 = 0; r0 < 256; r0 += 16) {
        int rr = r0 + trr;
        float2 v = base[(long)rr * 256 + cc];
        int br = __brev(rr) >> 24;
        re[br][cc] = v.x; im[br][cc] = v.y;
    }
    __syncthreads();
    for (int s = 1; s <= 8; s++) {
        const int half = 1 << (s - 1);
        #pragma unroll
        for (int q = 0; q < 8; q++) {
            const int bf  = trr + (q << 4);       // 0..127
            const int blk = bf >> (s - 1);
            const int pos = bf & (half - 1);
            const int i = (blk << s) + pos;
            const int j = i + half;
            const float ang = -6.28318530717958647f * (float)pos / (float)(1 << s);
            const float wc = cosf(ang), wsn = sinf(ang);
            float xr = re[j][cc], xi = im[j][cc];
            float tr = wc * xr - wsn * xi;
            float ti = wc * xi + wsn * xr;
            float ur = re[i][cc], ui = im[i][cc];
            re[i][cc] = ur + tr; im[i][cc] = ui + ti;
            re[j][cc] = ur - tr; im[j][cc] = ui - ti;
        }
        __syncthreads();
    }
    float* ob = outr + (long)img * 65536 + c0;
    for (int r0 = 0; r0 < 256; r0 += 16) {
        int rr = r0 + trr;
        ob[(long)rr * 256 + cc] = re[rr][cc] * scale;
    }
}

// Depthwise 3x3, SAME zero padding. One block per (b,c,row).
__global__ __launch_bounds__(256) void dwconv3x3k(
    const float* __restrict__ in, const float* __restrict__ wt, float* __restrict__ outp)
{
    const long bcr = blockIdx.x;
    const int r  = (int)(bcr & 255);
    const long bc = bcr >> 8;
    const int c  = (int)(bc % C_);
    const float* w  = wt + c * 9;
    const float* ip = in + bc * 65536;
    const int x = threadIdx.x;
    float acc = 0.f;
    #pragma unroll
    for (int dy = -1; dy <= 1; dy++) {
        int yy = r + dy;
        if (yy < 0 || yy > 255) continue;
        #pragma unroll
        for (int dx = -1; dx <= 1; dx++) {
            int xx = x + dx;
            if (xx < 0 || xx > 255) continue;
            acc += w[(dy + 1) * 3 + (dx + 1)] * ip[yy * 256 + xx];
        }
    }
    outp[bc * 65536 + r * 256 + x] = acc;
}

// ---------------------------------------------------------------------------
extern "C" void kernel_launch(void* const* d_in, const int* in_sizes, int n_in,
                              void* d_out, int out_size, void* d_ws, size_t ws_size,
                              hipStream_t stream)
{
    const float* x       = (const float*)d_in[0];
    const float* temp    = (const float*)d_in[1];
    const float* w_qkv   = (const float*)d_in[2];
    const float* w_proj  = (const float*)d_in[3];
    const float* w_kv    = (const float*)d_in[4];
    const float* w_q1    = (const float*)d_in[5];
    const float* w_q2    = (const float*)d_in[6];
    const float* w_kvc   = (const float*)d_in[7];
    const float* w_projf = (const float*)d_in[8];
    float* outp = (float*)d_out;

    float* ws = (float*)d_ws;
    const long PL = (long)C_ * NSP;            // per-batch channel plane block
    float* qkv   = ws;                         // B * 3C * N
    float* bout  = qkv   + 3L * B_ * PL;       // B * C * N
    float* bout2 = bout  + (long)B_ * PL;
    float* cplx  = bout2 + (long)B_ * PL;      // B * C * N complex
    float* r1    = cplx  + 2L * B_ * PL;
    float* mid   = r1    + (long)B_ * PL;
    float* qf    = mid   + (long)B_ * PL;
    float* kvb   = qf    + (long)B_ * PL;
    float* kvc   = kvb   + (long)B_ * PL;
    float* ctx   = kvc   + (long)B_ * PL;      // 16 * 48*48
    float* attnS = ctx   + 16L * 2304;
    float* attnf = attnS + 16L * 2304;
    float* norms = attnf + 16L * 2304;         // 768
    size_t need = (size_t)((norms + 768) - ws) * sizeof(float);
    if (ws_size < need) return;

    const dim3 gblk(128);
    const long DN = (long)D_ * NSP;

    // qkv = w_qkv @ x  (M=576, K=192, per batch)
    gemm_wmma<<<dim3(512, 9, B_), gblk, 0, stream>>>(qkv, w_qkv, x, 3 * C_, C_,
        0L, 0L, PL, 0L, 3L * PL, 0L, 1, 0, 0);
    // softmax over N for q,k rows
    softmax_rows_qk<<<dim3(B_ * 2 * C_), dim3(256), 0, stream>>>(qkv);
    // context[d,e] = sum_n k[d,n] v[e,n]
    (void)hipMemsetAsync(ctx, 0, 16L * 2304 * sizeof(float), stream);
    dotk_wmma<<<dim3(3, 3, 16 * 64), dim3(32), 0, stream>>>(ctx,
        qkv + (long)C_ * NSP, qkv + 2L * C_ * NSP,
        3L * PL, DN, 3L * PL, DN, NH_, 64);
    // out[e,n] = sum_d context[d,e] q[d,n]   (transA)
    gemm_wmma<<<dim3(512, 1, 16), gblk, 0, stream>>>(bout, ctx, qkv, D_, D_,
        (long)NH_ * D_ * D_, (long)D_ * D_, 3L * PL, DN, PL, DN, NH_, 1, 0);
    // out2 = w_proj @ out
    gemm_wmma<<<dim3(512, 3, B_), gblk, 0, stream>>>(bout2, w_proj, bout, C_, C_,
        0L, 0L, PL, 0L, PL, 0L, 1, 0, 0);
    // x_fft = real(FFT2(x))
    fft_rows<<<dim3(B_ * C_ * HW_ / 2), dim3(256), 0, stream>>>(x, (float2*)cplx);
    fft_cols<<<dim3(B_ * C_, 16), dim3(256), 0, stream>>>((const float2*)cplx, r1, 1.0f);
    // mid = gelu(w_q1 @ x_fft)
    gemm_wmma<<<dim3(512, 3, B_), gblk, 0, stream>>>(mid, w_q1, r1, C_, C_,
        0L, 0L, PL, 0L, PL, 0L, 1, 0, 1);
    // z = w_q2 @ mid
    gemm_wmma<<<dim3(512, 3, B_), gblk, 0, stream>>>(r1, w_q2, mid, C_, C_,
        0L, 0L, PL, 0L, PL, 0L, 1, 0, 0);
    // qf = real(IFFT2(z)) = real(FFT2(z)) / 65536   (z real)
    fft_rows<<<dim3(B_ * C_ * HW_ / 2), dim3(256), 0, stream>>>(r1, (float2*)cplx);
    fft_cols<<<dim3(B_ * C_, 16), dim3(256), 0, stream>>>((const float2*)cplx, qf,
                                                          1.0f / 65536.0f);
    // ||qf_d||
    rownorm<<<dim3(B_ * C_), dim3(256), 0, stream>>>(qf, norms);
    // vf_pre = w_kv[C:2C] @ out2   (kf half is dead code in the reference)
    gemm_wmma<<<dim3(512, 3, B_), gblk, 0, stream>>>(kvb, w_kv + (long)C_ * C_,
        bout2, C_, C_, 0L, 0L, PL, 0L, PL, 0L, 1, 0, 0);
    // vf = dwconv3x3(vf_pre)  (second-half depthwise weights)
    dwconv3x3k<<<dim3(B_ * C_ * HW_), dim3(256), 0, stream>>>(kvb,
        w_kvc + (long)C_ * 9, kvc);
    // attnS[d,e] = sum_n qf[d,n] k[e,n]
    (void)hipMemsetAsync(attnS, 0, 16L * 2304 * sizeof(float), stream);
    dotk_wmma<<<dim3(3, 3, 16 * 64), dim3(32), 0, stream>>>(attnS,
        qf, qkv + (long)C_ * NSP, PL, DN, 3L * PL, DN, NH_, 64);
    // attnf = softmax_e(attnS * T / ||qf_d||)
    attn_softmax<<<dim3(16 * D_), dim3(64), 0, stream>>>(attnS, norms, temp, attnf);
    // outf[d,n] = sum_e attnf[d,e] vf[e,n]
    gemm_wmma<<<dim3(512, 1, 16), gblk, 0, stream>>>(bout, attnf, kvc, D_, D_,
        (long)NH_ * D_ * D_, (long)D_ * D_, PL, DN, PL, DN, NH_, 0, 0);
    // final = w_projf @ outf  -> d_out
    gemm_wmma<<<dim3(512, 3, B_), gblk, 0, stream>>>(outp, w_projf, bout, C_, C_,
        0L, 0L, PL, 0L, PL, 0L, 1, 0, 0);
}